// GBST_7404523618863
// MI455X (gfx1250) — compile-verified
//
#include <hip/hip_runtime.h>
#include <math.h>

typedef __attribute__((ext_vector_type(2))) float v2f;
typedef __attribute__((ext_vector_type(8))) float v8f;

#define DIMD 512
#define SEQ  2048
#define BQ   8
#define SP   2052
#define NB2  1026
#define NB3  684
#define NB4  513
#define MAXNEG (-3.402823466e38f)

// ---------------- Kernel 1: embedding gather + depthwise conv (width 4) ----------------
__global__ void k_embed_conv(const int* __restrict__ x, const float* __restrict__ emb,
                             const float* __restrict__ dww, const float* __restrict__ dwb,
                             float* __restrict__ hconv) {
    int tid = blockIdx.x * blockDim.x + threadIdx.x;   // exactly B*S*D threads
    int d   = tid & (DIMD - 1);
    int pos = tid >> 9;                                // b*SEQ + s
    int b   = pos >> 11;
    int s   = pos & (SEQ - 1);
    float acc = dwb[d];
#pragma unroll
    for (int k = 0; k < 4; ++k) {
        int sp = s + k;
        if (sp < SEQ) {
            int tok = x[b * SEQ + sp];
            acc += emb[tok * DIMD + d] * dww[k * DIMD + d];
        }
    }
    hconv[pos * DIMD + d] = acc;
}

// ---------------- Kernel 2: pointwise GEMM via native f32 WMMA ----------------
// C[M,N] = A[M,K] * B[K,N] + bias[N];  M=16384, N=K=512.
// Block = 128 threads (4 waves). Block tile 64x64, wave tile 16x64 (4 accumulators).
__global__ __launch_bounds__(128) void k_gemm_wmma(const float* __restrict__ A,
                                                   const float* __restrict__ Bm,
                                                   const float* __restrict__ bias,
                                                   float* __restrict__ C) {
    __shared__ float lA[16 * 65];   // [k][m], padded stride 65
    __shared__ float lB[16 * 65];   // [k][n], padded stride 65
    const int t    = threadIdx.x;
    const int lane = t & 31;
    const int w    = t >> 5;
    const int mr   = lane & 15;
    const int kh   = (lane >> 4) << 1;      // lanes 16-31 hold K+2,K+3 halves
    const int blockM = blockIdx.x * 64;
    const int blockN = blockIdx.y * 64;
    v8f acc[4] = {};

    for (int kk = 0; kk < DIMD; kk += 16) {
        __syncthreads();
#pragma unroll
        for (int i = 0; i < 8; ++i) {
            int e  = i * 128 + t;
            int r  = e >> 4, c  = e & 15;               // A: 64 rows x 16 k
            lA[c * 65 + r] = A[(blockM + r) * DIMD + kk + c];
            int rb = e >> 6, cb = e & 63;               // B: 16 k x 64 n
            lB[rb * 65 + cb] = Bm[(kk + rb) * DIMD + blockN + cb];
        }
        __syncthreads();
#pragma unroll
        for (int k4 = 0; k4 < 4; ++k4) {
            int k0 = k4 * 4 + kh;
            v2f a;
            a.x = lA[k0 * 65 + w * 16 + mr];
            a.y = lA[(k0 + 1) * 65 + w * 16 + mr];
#pragma unroll
            for (int nt = 0; nt < 4; ++nt) {
                v2f bb;
                bb.x = lB[k0 * 65 + nt * 16 + mr];
                bb.y = lB[(k0 + 1) * 65 + nt * 16 + mr];
                acc[nt] = __builtin_amdgcn_wmma_f32_16x16x4_f32(
                    false, a, false, bb, (short)0, acc[nt], false, false);
            }
        }
    }

    // C/D layout: VGPR i, lanes 0-15 -> M=i, N=lane; lanes 16-31 -> M=i+8, N=lane-16
    int rowBase = blockM + w * 16 + ((lane >> 4) << 3);
#pragma unroll
    for (int nt = 0; nt < 4; ++nt) {
        int col  = blockN + nt * 16 + mr;
        float bv = bias[col];
#pragma unroll
        for (int i = 0; i < 8; ++i)
            C[(rowBase + i) * DIMD + col] = acc[nt][i] + bv;
    }
}

// ---------------- Kernel 3: masked block means for bsz in {2,3,4} ----------------
__global__ void k_block_mean(const float* __restrict__ h2, const int* __restrict__ mask,
                             float* __restrict__ r, int* __restrict__ anyf,
                             int bsz, int nb) {
    long tid = (long)blockIdx.x * blockDim.x + threadIdx.x;
    long total = (long)BQ * nb * DIMD;
    if (tid >= total) return;
    int  d   = (int)(tid & (DIMD - 1));
    int  row = (int)(tid >> 9);
    int  b   = row / nb;
    int  j   = row - b * nb;
    int  s0  = j * bsz;
    float sum = 0.f; int cnt = 0;
    for (int q = 0; q < bsz; ++q) {
        int s = s0 + q;
        if (s < SEQ && mask[b * SEQ + s]) { sum += h2[(b * SEQ + s) * DIMD + d]; cnt++; }
    }
    r[row * DIMD + d] = cnt ? sum / (float)cnt : 0.f;
    if (d == 0) anyf[row] = (cnt > 0);
}

// ---------------- Kernel 4: row . score_w + score_b ----------------
__global__ __launch_bounds__(128) void k_dot(const float* __restrict__ src,
                                             const float* __restrict__ sw,
                                             const float* __restrict__ sb,
                                             float* __restrict__ dst, int rows) {
    __shared__ float red[128];
    int row = blockIdx.x;
    if (row >= rows) return;
    int t = threadIdx.x;
    float p = 0.f;
    for (int d = t; d < DIMD; d += 128) p += src[row * DIMD + d] * sw[d];
    red[t] = p;
    __syncthreads();
    for (int off = 64; off > 0; off >>= 1) {
        if (t < off) red[t] += red[t + off];
        __syncthreads();
    }
    if (t == 0) dst[row] = red[0] + sb[0];
}

// ---------------- Kernel 5: 4-way masked softmax per position + row mask ----------------
__global__ void k_softmax4(const int* __restrict__ mask,
                           const float* __restrict__ sc1, const float* __restrict__ sc2,
                           const float* __restrict__ sc3, const float* __restrict__ sc4,
                           const int* __restrict__ any2, const int* __restrict__ any3,
                           const int* __restrict__ any4,
                           float4* __restrict__ probs, int* __restrict__ mrow) {
    int u = blockIdx.x * blockDim.x + threadIdx.x;
    if (u >= BQ * SP) return;
    int b = u / SP, s = u - b * SP;
    int mp = (s < SEQ) && mask[b * SEQ + s];
    mrow[u] = mp;
    float v0 = mp ? sc1[b * SEQ + s] : MAXNEG;
    int j2 = s >> 1; float v1 = any2[b * NB2 + j2] ? sc2[b * NB2 + j2] : MAXNEG;
    int j3 = s / 3;  float v2 = any3[b * NB3 + j3] ? sc3[b * NB3 + j3] : MAXNEG;
    int j4 = s >> 2; float v3 = any4[b * NB4 + j4] ? sc4[b * NB4 + j4] : MAXNEG;
    float mx = fmaxf(fmaxf(v0, v1), fmaxf(v2, v3));
    float e0 = __expf(v0 - mx), e1 = __expf(v1 - mx), e2 = __expf(v2 - mx), e3 = __expf(v3 - mx);
    float Z = e0 + e1 + e2 + e3;
    probs[u] = make_float4(e0 / Z, e1 / Z, e2 / Z, e3 / Z);
}

// ---------------- Kernel 6: fused row attention (K=4), online softmax over j ----------------
__global__ __launch_bounds__(128) void k_attn(const float4* __restrict__ probs,
                                              const int* __restrict__ mrow,
                                              float4* __restrict__ osc) {
    __shared__ float sm[128 * 6];
    int u = blockIdx.x;           // b*SP + i
    int b = u / SP;
    int t = threadIdx.x;
    if (!mrow[u]) { if (t == 0) osc[u] = make_float4(0.f, 0.f, 0.f, 0.f); return; }
    float4 pi = probs[u];
    const float4* pb = probs + b * SP;
    const int*    mb = mrow + b * SP;
    float ml = MAXNEG, Z = 0.f, n0 = 0.f, n1 = 0.f, n2 = 0.f, n3 = 0.f;
    for (int j = t; j < SP; j += 128) {
        if (!mb[j]) continue;
        float4 pj = pb[j];
        float l = pi.x * pj.x + pi.y * pj.y + pi.z * pj.z + pi.w * pj.w;
        if (l > ml) {
            float f = __expf(ml - l);
            Z *= f; n0 *= f; n1 *= f; n2 *= f; n3 *= f;
            ml = l;
            Z += 1.f; n0 += pj.x; n1 += pj.y; n2 += pj.z; n3 += pj.w;
        } else {
            float e = __expf(l - ml);
            Z += e; n0 += e * pj.x; n1 += e * pj.y; n2 += e * pj.z; n3 += e * pj.w;
        }
    }
    sm[t*6+0] = ml; sm[t*6+1] = Z; sm[t*6+2] = n0; sm[t*6+3] = n1; sm[t*6+4] = n2; sm[t*6+5] = n3;
    __syncthreads();
    if (t == 0) {
        float gm = MAXNEG;
        for (int q = 0; q < 128; ++q) gm = fmaxf(gm, sm[q*6]);
        float Zt = 0.f, a0 = 0.f, a1 = 0.f, a2 = 0.f, a3 = 0.f;
        for (int q = 0; q < 128; ++q) {
            float f = __expf(sm[q*6] - gm);
            Zt += sm[q*6+1] * f; a0 += sm[q*6+2] * f; a1 += sm[q*6+3] * f;
            a2 += sm[q*6+4] * f; a3 += sm[q*6+5] * f;
        }
        float inv = 1.f / Zt;
        osc[u] = make_float4(a0 * inv, a1 * inv, a2 * inv, a3 * inv);
    }
}

// ---------------- Kernel 7: recombine 4 scales + DS=4 masked pooling + out_mask ----------------
__global__ void k_final(const float* __restrict__ h2, const float* __restrict__ r2,
                        const float* __restrict__ r3, const float* __restrict__ r4,
                        const float4* __restrict__ osc, const int* __restrict__ mask,
                        float* __restrict__ out, float* __restrict__ omask) {
    int tid  = blockIdx.x * blockDim.x + threadIdx.x;  // exactly B*512*D
    int d    = tid & (DIMD - 1);
    int rest = tid >> 9;
    int l    = rest & 511;
    int b    = rest >> 9;
    float sum = 0.f; int cnt = 0;
#pragma unroll
    for (int q = 0; q < 4; ++q) {
        int s = 4 * l + q;
        if (mask[b * SEQ + s]) {
            cnt++;
            float4 wv = osc[b * SP + s];
            sum += h2[(b * SEQ + s) * DIMD + d] * wv.x
                 + r2[(b * NB2 + (s >> 1)) * DIMD + d] * wv.y
                 + r3[(b * NB3 + (s / 3)) * DIMD + d] * wv.z
                 + r4[(b * NB4 + (s >> 2)) * DIMD + d] * wv.w;
        }
    }
    out[(b * 512 + l) * DIMD + d] = cnt ? sum / (float)cnt : 0.f;
    if (d == 0) omask[b * 512 + l] = cnt ? 1.0f : 0.0f;
}

extern "C" void kernel_launch(void* const* d_in, const int* in_sizes, int n_in,
                              void* d_out, int out_size, void* d_ws, size_t ws_size,
                              hipStream_t stream) {
    (void)in_sizes; (void)n_in; (void)out_size; (void)ws_size;
    const int*   x    = (const int*)d_in[0];
    const int*   mask = (const int*)d_in[1];
    const float* emb  = (const float*)d_in[2];
    const float* dww  = (const float*)d_in[3];
    const float* dwb  = (const float*)d_in[4];
    const float* pww  = (const float*)d_in[5];
    const float* pwb  = (const float*)d_in[6];
    const float* scw  = (const float*)d_in[7];
    const float* scb  = (const float*)d_in[8];

    char* ws = (char*)d_ws;
    size_t off = 0;
    auto alloc = [&](size_t bytes) -> void* {
        void* p = ws + off;
        off += (bytes + 255) & ~(size_t)255;
        return p;
    };

    const size_t MROWS = (size_t)BQ * SEQ;                 // 16384
    float* hconv = (float*)alloc(MROWS * DIMD * 4);        // dead after GEMM -> reused for r2/r3
    float* h2    = (float*)alloc(MROWS * DIMD * 4);
    float* r4    = (float*)alloc((size_t)BQ * NB4 * DIMD * 4);
    int*   any2  = (int*)  alloc((size_t)BQ * NB2 * 4);
    int*   any3  = (int*)  alloc((size_t)BQ * NB3 * 4);
    int*   any4  = (int*)  alloc((size_t)BQ * NB4 * 4);
    float* sc1   = (float*)alloc((size_t)BQ * SEQ * 4);
    float* sc2   = (float*)alloc((size_t)BQ * NB2 * 4);
    float* sc3   = (float*)alloc((size_t)BQ * NB3 * 4);
    float* sc4   = (float*)alloc((size_t)BQ * NB4 * 4);
    float4* probs = (float4*)alloc((size_t)BQ * SP * 16);
    int*    mrow  = (int*)   alloc((size_t)BQ * SP * 4);
    float4* osc   = (float4*)alloc((size_t)BQ * SP * 16);
    // r2/r3 overlap the dead hconv region (16.8MB + 11.2MB <= 33.5MB)
    float* r2 = hconv;
    float* r3 = hconv + (size_t)BQ * NB2 * DIMD;

    // 1) embed + depthwise conv
    k_embed_conv<<<(BQ * SEQ * DIMD) / 256, 256, 0, stream>>>(x, emb, dww, dwb, hconv);
    // 2) pointwise GEMM (WMMA f32)
    dim3 g2(BQ * SEQ / 64, DIMD / 64);
    k_gemm_wmma<<<g2, 128, 0, stream>>>(hconv, pww, pwb, h2);
    // 3) multi-scale block means
    k_block_mean<<<(BQ * NB2 * DIMD + 255) / 256, 256, 0, stream>>>(h2, mask, r2, any2, 2, NB2);
    k_block_mean<<<(BQ * NB3 * DIMD + 255) / 256, 256, 0, stream>>>(h2, mask, r3, any3, 3, NB3);
    k_block_mean<<<(BQ * NB4 * DIMD + 255) / 256, 256, 0, stream>>>(h2, mask, r4, any4, 4, NB4);
    // 4) scores per scale
    k_dot<<<BQ * SEQ, 128, 0, stream>>>(h2, scw, scb, sc1, BQ * SEQ);
    k_dot<<<BQ * NB2, 128, 0, stream>>>(r2, scw, scb, sc2, BQ * NB2);
    k_dot<<<BQ * NB3, 128, 0, stream>>>(r3, scw, scb, sc3, BQ * NB3);
    k_dot<<<BQ * NB4, 128, 0, stream>>>(r4, scw, scb, sc4, BQ * NB4);
    // 5) per-position 4-way softmax
    k_softmax4<<<(BQ * SP + 255) / 256, 256, 0, stream>>>(mask, sc1, sc2, sc3, sc4,
                                                          any2, any3, any4, probs, mrow);
    // 6) fused attention over sequence (online softmax, K=4)
    k_attn<<<BQ * SP, 128, 0, stream>>>(probs, mrow, osc);
    // 7) recombine + DS pooling + out mask
    float* outp  = (float*)d_out;
    float* omask = outp + (size_t)BQ * 512 * DIMD;
    k_final<<<(BQ * 512 * DIMD) / 256, 256, 0, stream>>>(h2, r2, r3, r4, osc, mask, outp, omask);
}